// VolumeRenderer_15650860827223
// MI455X (gfx1250) — compile-verified
//
#include <hip/hip_runtime.h>

// Volume rendering composite: exclusive prefix-product scan + weighted RGB sum.
// Memory-bound (~643 MiB @ 23.3 TB/s => ~29 us floor). One wave32 per ray,
// lane l owns samples [4l, 4l+3]. Alpha tile staged through LDS with the
// CDNA5 async-to-LDS engine; rgbs/weights streamed with non-temporal b128 ops.

typedef float floatx4 __attribute__((ext_vector_type(4)));
typedef int   intx4   __attribute__((ext_vector_type(4)));

typedef __attribute__((address_space(1))) intx4* global_b128_ptr;
typedef __attribute__((address_space(3))) intx4* lds_b128_ptr;

#define SAMPLES 128
#define WAVES_PER_BLOCK 8

#if defined(__gfx1250__) && __has_builtin(__builtin_amdgcn_global_load_async_to_lds_b128)
#define VR_ASYNC 1
#else
#define VR_ASYNC 0
#endif

__global__ __launch_bounds__(WAVES_PER_BLOCK * 32)
void vr_composite_kernel(const float* __restrict__ alpha,
                         const float* __restrict__ rgbs,
                         float* __restrict__ out_rgb,
                         float* __restrict__ out_w,
                         int n_rays)
{
    const int lane = threadIdx.x & 31;
    const int wv   = threadIdx.x >> 5;
    const int ray  = blockIdx.x * WAVES_PER_BLOCK + wv;
    if (ray >= n_rays) return;

    __shared__ float s_alpha[WAVES_PER_BLOCK * SAMPLES];

    const size_t abase = (size_t)ray * SAMPLES + (size_t)lane * 4;

    floatx4 a;
#if VR_ASYNC
    {
        float* lds_dst = &s_alpha[wv * SAMPLES + lane * 4];
        // Per-lane 16B async copy global -> LDS (ASYNCcnt path, CDNA5).
        __builtin_amdgcn_global_load_async_to_lds_b128(
            (global_b128_ptr)(alpha + abase),
            (lds_b128_ptr)lds_dst,
            0, 0);
        asm volatile("s_wait_asynccnt 0" ::: "memory");
        a = *(const floatx4*)lds_dst;   // ds_load_b128
    }
#else
    a = __builtin_nontemporal_load((const floatx4*)(alpha + abase));
#endif

    // f_i = (1 - alpha_i) + eps, exclusive prefixes within the lane's 4 samples.
    const float eps = 1e-10f;
    const float f0 = 1.0f - a[0] + eps;
    const float f1 = 1.0f - a[1] + eps;
    const float f2 = 1.0f - a[2] + eps;
    const float f3 = 1.0f - a[3] + eps;
    const float e1 = f0;
    const float e2 = f0 * f1;
    const float e3 = e2 * f2;
    const float tot = e3 * f3;

    // Wave32 inclusive scan-product of per-lane totals, then shift to exclusive.
    float scan = tot;
#pragma unroll
    for (int d = 1; d < 32; d <<= 1) {
        float up = __shfl_up(scan, d, 32);
        if (lane >= d) scan *= up;
    }
    float P = __shfl_up(scan, 1, 32);
    if (lane == 0) P = 1.0f;

    // weights_i = alpha_i * prod_{j<i} f_j
    floatx4 w;
    w[0] = a[0] * P;
    w[1] = a[1] * (P * e1);
    w[2] = a[2] * (P * e2);
    w[3] = a[3] * (P * e3);

    __builtin_nontemporal_store(w, (floatx4*)(out_w + abase));

    // rgbs: 12 contiguous floats per lane (48B, 16B aligned) = 3 x b128 NT loads.
    const float* rp = rgbs + (size_t)ray * (SAMPLES * 3) + (size_t)lane * 12;
    const floatx4 q0 = __builtin_nontemporal_load((const floatx4*)(rp));
    const floatx4 q1 = __builtin_nontemporal_load((const floatx4*)(rp + 4));
    const floatx4 q2 = __builtin_nontemporal_load((const floatx4*)(rp + 8));
    // q0={r0,g0,b0,r1} q1={g1,b1,r2,g2} q2={b2,r3,g3,b3}
    float r = w[0]*q0[0] + w[1]*q0[3] + w[2]*q1[2] + w[3]*q2[1];
    float g = w[0]*q0[1] + w[1]*q1[0] + w[2]*q1[3] + w[3]*q2[2];
    float b = w[0]*q0[2] + w[1]*q1[1] + w[2]*q2[0] + w[3]*q2[3];

#pragma unroll
    for (int off = 16; off > 0; off >>= 1) {
        r += __shfl_down(r, off, 32);
        g += __shfl_down(g, off, 32);
        b += __shfl_down(b, off, 32);
    }
    if (lane == 0) {
        float* o = out_rgb + (size_t)ray * 3;
        o[0] = r; o[1] = g; o[2] = b;
    }
}

extern "C" void kernel_launch(void* const* d_in, const int* in_sizes, int n_in,
                              void* d_out, int out_size, void* d_ws, size_t ws_size,
                              hipStream_t stream) {
    const float* alpha = (const float*)d_in[0];     // [R, 128]
    const float* rgbs  = (const float*)d_in[1];     // [R, 128, 3]
    const int n_rays = in_sizes[0] / SAMPLES;

    float* out_rgb = (float*)d_out;                      // [R, 3]
    float* out_w   = out_rgb + (size_t)n_rays * 3;       // [R, 128]

    const int blocks = (n_rays + WAVES_PER_BLOCK - 1) / WAVES_PER_BLOCK;
    vr_composite_kernel<<<blocks, WAVES_PER_BLOCK * 32, 0, stream>>>(
        alpha, rgbs, out_rgb, out_w, n_rays);
}